// getDataTerm_AhAx_12137577578890
// MI455X (gfx1250) — compile-verified
//
#include <hip/hip_runtime.h>

// ---------------------------------------------------------------------------
// MRI SENSE forward+adjoint (getDataTerm A^H A x) on MI455X / gfx1250.
//
// Every centered 2D FFT/IFFT is cast as Fc·X·Fc with Fc the centered, scaled
// 256x256 DFT matrix (Fc symmetric; inverse = conj(Fc)).  All transforms run
// through a complex f32 GEMM built on V_WMMA_F32_16X16X4_F32.
//
// Workspace layout (floats):
//   [OFF_F  ] Fc re/im planes                 :   131072
//   [OFF_M1 ] M1[seg][coil][re|im][256*256]   : 16777216  (also aliased as M2)
//   [OFF_KS ] kspace[coil][re|im][256*256]    :  4194304
//   [OFF_Z  ] Z[coil][re|im][256*256]         :  4194304
//   total ~101.2 MB  (reference's own intermediates are >500 MB, so ws is
//   expected to cover this)
// ---------------------------------------------------------------------------

typedef __attribute__((ext_vector_type(2))) float v2f;
typedef __attribute__((ext_vector_type(8))) float v8f;

#define NPIX   65536
#define RDIM   256
#define CDIM   256
#define NB     4
#define NCOIL  32
#define NSEG   4
#define NPOL   2
#define XDIM   276
#define CROPO  10

#define OFF_F   0u
#define OFF_M1  131072u
#define OFF_KS  (OFF_M1 + 16777216u)
#define OFF_Z   (OFF_KS + 4194304u)

__device__ __forceinline__ v8f wmma_f32(v2f a, v2f b, v8f c) {
  // D = A(16x4) * B(4x16) + C(16x16), all f32.  f32 WMMA has no A/B NEG
  // modifiers (ISA 7.12: NEG = CNeg only), so conj/negation is done on the
  // fragments explicitly.
  return __builtin_amdgcn_wmma_f32_16x16x4_f32(
      false, a, false, b, (short)0, c, false, false);
}

// ---- centered, scaled DFT matrix: Fc[u,r] = exp(-2pi i (u-128)(r-128)/256)/16
__global__ void init_dft(float* __restrict__ F) {
  int idx = blockIdx.x * blockDim.x + threadIdx.x;
  if (idx >= NPIX) return;
  int u = idx >> 8, r = idx & 255;
  // integer phase reduction mod 256 keeps sincos argument small & exact
  int p = ((u - 128) * (r - 128)) & 255;
  float ang = -6.2831853071795864769f * (float)p * (1.0f / 256.0f);
  float s, c;
  sincosf(ang, &s, &c);
  F[idx]        = c * 0.0625f;
  F[NPIX + idx] = s * 0.0625f;
}

__global__ void zero_f32(float* __restrict__ p, int n) {
  int i = blockIdx.x * blockDim.x + threadIdx.x;
  if (i < n) p[i] = 0.0f;
}

// ---------------------------------------------------------------------------
// Fused complex GEMM, 64x64 workgroup tile, 128 threads (4 wave32s).
// MODE 0 (FWD1): M1[seg,coil]  = Fc        * (img*wmap*csm)        z = seg*32+coil
// MODE 1 (FWD2): kspace[coil]  = sum_seg (M1[seg,coil]*Fc) .* mask z = coil
// MODE 2 (ADJ1): M2[coil]      = conj(Fc)  * (kspace[coil].*mask)  z = coil
// MODE 3 (ADJ2): Z[coil]       = (M2[coil]*conj(Fc)) .* conj(wmap).*conj(csm)
// ---------------------------------------------------------------------------
template <int MODE>
__global__ __launch_bounds__(128)
void gemm_mri(const float* __restrict__ xre, const float* __restrict__ xim,
              const float* __restrict__ csre, const float* __restrict__ csim,
              const float* __restrict__ wre,  const float* __restrict__ wim,
              const float* __restrict__ msk,
              float* __restrict__ ws, int b, int pol, int segArg) {
  const float* F  = ws + OFF_F;
  float* M1       = ws + OFF_M1;   // M2 aliases this region in the adjoint
  float* KS       = ws + OFF_KS;
  float* Zb       = ws + OFF_Z;

  __shared__ float As[2][64][17];  // pitch 17: conflict-free column reads
  __shared__ float Bs[2][16][64];

  const int tid  = threadIdx.x;
  const int lane = tid & 31;
  const int wv   = tid >> 5;
  const int qr   = (wv >> 1) * 32;        // wave quadrant inside 64x64 tile
  const int qc   = (wv & 1) * 32;
  const int rowBase = blockIdx.y * 64;
  const int colBase = blockIdx.x * 64;

  int coil, seg;
  if (MODE == 0) { seg = blockIdx.z >> 5; coil = blockIdx.z & 31; }
  else           { coil = blockIdx.z;     seg  = segArg; }

  const int am  = lane & 15;              // A row / B,C column lane index
  const int akb = (lane >> 4) * 2;        // A/B k sub-offset per ISA layout

  const v8f vzero = {0.f, 0.f, 0.f, 0.f, 0.f, 0.f, 0.f, 0.f};

  float kaccR[2][2][8], kaccI[2][2][8];
  if constexpr (MODE == 1) {
    for (int ti = 0; ti < 2; ++ti)
      for (int tj = 0; tj < 2; ++tj)
        for (int j = 0; j < 8; ++j) { kaccR[ti][tj][j] = 0.f; kaccI[ti][tj][j] = 0.f; }
  }

  const int nseg = (MODE == 1) ? NSEG : 1;
  for (int sg = 0; sg < nseg; ++sg) {
    const int s_eff = (MODE == 1) ? sg : seg;

    v8f accR[2][2], accI[2][2];
    for (int ti = 0; ti < 2; ++ti)
      for (int tj = 0; tj < 2; ++tj) { accR[ti][tj] = vzero; accI[ti][tj] = vzero; }

    const float* Asrc = nullptr;
    if constexpr (MODE == 1) Asrc = M1 + (size_t)((s_eff * NCOIL + coil) * 2) * NPIX;
    if constexpr (MODE == 3) Asrc = M1 + (size_t)(coil * 2) * NPIX;  // == M2

    for (int kt = 0; kt < 256; kt += 16) {
      // -------- stage A tile (64 rows x 16 k, complex) --------
      for (int i = 0; i < 8; ++i) {
        int e = i * 128 + tid;
        int ar_ = e >> 4, ac_ = e & 15;
        int gr = rowBase + ar_, gk = kt + ac_;
        float vr, vi;
        if constexpr (MODE == 0)      { vr = F[gr * 256 + gk]; vi =  F[NPIX + gr * 256 + gk]; }
        else if constexpr (MODE == 2) { vr = F[gr * 256 + gk]; vi = -F[NPIX + gr * 256 + gk]; }
        else                          { vr = Asrc[gr * 256 + gk]; vi = Asrc[NPIX + gr * 256 + gk]; }
        As[0][ar_][ac_] = vr;
        As[1][ar_][ac_] = vi;
      }
      // -------- stage B tile (16 k x 64 cols, complex, fused prologue) -----
      for (int i = 0; i < 8; ++i) {
        int e = i * 128 + tid;
        int br_ = e >> 6, bc_ = e & 63;
        int gk = kt + br_, gc = colBase + bc_;
        float vr, vi;
        if constexpr (MODE == 0) {
          // T[r,c] = img * wmap * csm   (img = center-crop of x)
          int xi = ((b * XDIM + gk + CROPO) * XDIM + gc + CROPO) * NPOL + pol;
          float ir = xre[xi], ii = xim[xi];
          int wiI = (((b * RDIM + gk) * CDIM + gc) * NPOL + pol) * NSEG + s_eff;
          float wr_ = wre[wiI], wi_ = wim[wiI];
          int ci = ((b * RDIM + gk) * CDIM + gc) * NCOIL + coil;
          float sr = csre[ci], si = csim[ci];
          float tr = ir * wr_ - ii * wi_;
          float tv = ir * wi_ + ii * wr_;
          vr = tr * sr - tv * si;
          vi = tr * si + tv * sr;
        } else if constexpr (MODE == 1) {
          vr = F[gk * 256 + gc]; vi = F[NPIX + gk * 256 + gc];
        } else if constexpr (MODE == 2) {
          int mi = (((b * RDIM + gk) * CDIM + gc) * NPOL + pol) * NSEG + s_eff;
          float m = msk[mi];
          const float* ksb = KS + (size_t)(coil * 2) * NPIX;
          vr = ksb[gk * 256 + gc] * m;
          vi = ksb[NPIX + gk * 256 + gc] * m;
        } else {
          vr = F[gk * 256 + gc]; vi = -F[NPIX + gk * 256 + gc];  // conj(Fc)
        }
        Bs[0][br_][bc_] = vr;
        Bs[1][br_][bc_] = vi;
      }
      __syncthreads();

      // -------- WMMA micro-kernel: 4 real chains per complex tile ---------
      for (int kk = 0; kk < 16; kk += 4) {
        v2f ar2[2], ai2[2], ain2[2], br2[2], bi2[2];
        for (int t2 = 0; t2 < 2; ++t2) {
          int rr = qr + t2 * 16 + am;
          ar2[t2].x = As[0][rr][kk + akb];     ar2[t2].y = As[0][rr][kk + akb + 1];
          ai2[t2].x = As[1][rr][kk + akb];     ai2[t2].y = As[1][rr][kk + akb + 1];
          ain2[t2]  = -ai2[t2];
        }
        for (int t2 = 0; t2 < 2; ++t2) {
          int nc = qc + t2 * 16 + am;
          br2[t2].x = Bs[0][kk + akb][nc];     br2[t2].y = Bs[0][kk + akb + 1][nc];
          bi2[t2].x = Bs[1][kk + akb][nc];     bi2[t2].y = Bs[1][kk + akb + 1][nc];
        }
        for (int ti = 0; ti < 2; ++ti)
          for (int tj = 0; tj < 2; ++tj) {
            accR[ti][tj] = wmma_f32(ar2[ti],  br2[tj], accR[ti][tj]);  // +ArBr
            accR[ti][tj] = wmma_f32(ain2[ti], bi2[tj], accR[ti][tj]);  // -AiBi
            accI[ti][tj] = wmma_f32(ar2[ti],  bi2[tj], accI[ti][tj]);  // +ArBi
            accI[ti][tj] = wmma_f32(ai2[ti],  br2[tj], accI[ti][tj]);  // +AiBr
          }
      }
      __syncthreads();
    }  // kt

    // -------- epilogue (element coords per ISA C-layout) --------
    for (int ti = 0; ti < 2; ++ti)
      for (int tj = 0; tj < 2; ++tj)
        for (int j = 0; j < 8; ++j) {
          int m = rowBase + qr + ti * 16 + (lane >> 4) * 8 + j;
          int n = colBase + qc + tj * 16 + am;
          float zr = accR[ti][tj][j], zi = accI[ti][tj][j];
          if constexpr (MODE == 0) {
            float* dst = M1 + (size_t)((s_eff * NCOIL + coil) * 2) * NPIX;
            dst[m * 256 + n]        = zr;
            dst[NPIX + m * 256 + n] = zi;
          } else if constexpr (MODE == 1) {
            float mv = msk[(((b * RDIM + m) * CDIM + n) * NPOL + pol) * NSEG + s_eff];
            kaccR[ti][tj][j] += zr * mv;
            kaccI[ti][tj][j] += zi * mv;
          } else if constexpr (MODE == 2) {
            float* dst = M1 + (size_t)(coil * 2) * NPIX;  // M2
            dst[m * 256 + n]        = zr;
            dst[NPIX + m * 256 + n] = zi;
          } else {
            // z * conj(wmap) * conj(csm)
            int wiI = (((b * RDIM + m) * CDIM + n) * NPOL + pol) * NSEG + seg;
            float wr_ = wre[wiI], wi_ = wim[wiI];
            int ci = ((b * RDIM + m) * CDIM + n) * NCOIL + coil;
            float sr = csre[ci], si = csim[ci];
            float t1r = zr * wr_ + zi * wi_;
            float t1i = zi * wr_ - zr * wi_;
            float fr = t1r * sr + t1i * si;
            float fi = t1i * sr - t1r * si;
            float* dst = Zb + (size_t)(coil * 2) * NPIX;
            dst[m * 256 + n]        = fr;
            dst[NPIX + m * 256 + n] = fi;
          }
        }
  }  // sg

  if constexpr (MODE == 1) {
    float* dst = KS + (size_t)(coil * 2) * NPIX;
    for (int ti = 0; ti < 2; ++ti)
      for (int tj = 0; tj < 2; ++tj)
        for (int j = 0; j < 8; ++j) {
          int m = rowBase + qr + ti * 16 + (lane >> 4) * 8 + j;
          int n = colBase + qc + tj * 16 + am;
          dst[m * 256 + n]        = kaccR[ti][tj][j];
          dst[NPIX + m * 256 + n] = kaccI[ti][tj][j];
        }
  }
}

// sum Z over the 32 coils (fixed order -> deterministic) and += into the
// zero-initialized, center-padded output [B,276,276,POL,2]
__global__ void reduce_out(const float* __restrict__ ws, float* __restrict__ out,
                           int b, int pol) {
  int idx = blockIdx.x * blockDim.x + threadIdx.x;
  if (idx >= NPIX) return;
  int r = idx >> 8, c = idx & 255;
  const float* Zb = ws + OFF_Z;
  float sr = 0.f, si = 0.f;
  for (int coil = 0; coil < NCOIL; ++coil) {
    sr += Zb[(size_t)(coil * 2) * NPIX + idx];
    si += Zb[(size_t)(coil * 2 + 1) * NPIX + idx];
  }
  int o = (((b * XDIM + r + CROPO) * XDIM + (c + CROPO)) * NPOL + pol) * 2;
  out[o]     += sr;
  out[o + 1] += si;
}

extern "C" void kernel_launch(void* const* d_in, const int* in_sizes, int n_in,
                              void* d_out, int out_size, void* d_ws, size_t ws_size,
                              hipStream_t stream) {
  (void)in_sizes; (void)n_in; (void)ws_size;  // requires ~101.2 MB workspace
  const float* xre  = (const float*)d_in[0];
  const float* xim  = (const float*)d_in[1];
  const float* csre = (const float*)d_in[2];
  const float* csim = (const float*)d_in[3];
  const float* wre  = (const float*)d_in[4];
  const float* wim  = (const float*)d_in[5];
  const float* msk  = (const float*)d_in[6];
  float* out = (float*)d_out;
  float* ws  = (float*)d_ws;

  zero_f32<<<(out_size + 255) / 256, 256, 0, stream>>>(out, out_size);
  init_dft<<<NPIX / 256, 256, 0, stream>>>(ws + OFF_F);

  dim3 blk(128);
  for (int b = 0; b < NB; ++b)
    for (int pol = 0; pol < NPOL; ++pol) {
      gemm_mri<0><<<dim3(4, 4, NSEG * NCOIL), blk, 0, stream>>>(
          xre, xim, csre, csim, wre, wim, msk, ws, b, pol, 0);
      gemm_mri<1><<<dim3(4, 4, NCOIL), blk, 0, stream>>>(
          xre, xim, csre, csim, wre, wim, msk, ws, b, pol, 0);
      for (int seg = 0; seg < NSEG; ++seg) {
        gemm_mri<2><<<dim3(4, 4, NCOIL), blk, 0, stream>>>(
            xre, xim, csre, csim, wre, wim, msk, ws, b, pol, seg);
        gemm_mri<3><<<dim3(4, 4, NCOIL), blk, 0, stream>>>(
            xre, xim, csre, csim, wre, wim, msk, ws, b, pol, seg);
        reduce_out<<<NPIX / 256, 256, 0, stream>>>(ws, out, b, pol);
      }
    }
}